// ApplyLtLinWeight1d_2422361555614
// MI455X (gfx1250) — compile-verified
//
#include <hip/hip_runtime.h>

// y[i] = x[i] < 0.5f ? 0.1f * x[i] : x[i]
// Streaming elementwise op: 512 MB HBM traffic, zero reuse -> bandwidth bound
// (floor ~23 us @ 23.3 TB/s). Full-tile kernel is branch-free so the 4x b128
// non-temporal loads can issue back-to-back per wave; tail kernel handles any
// remainder (unused for n = 2^26) so no exec-mask diamonds pollute the hot path.

typedef float v4f __attribute__((ext_vector_type(4)));

#define LT_W 0.5f
#define LIN_W 0.1f

__device__ __forceinline__ float apply_elem(float v) {
    return v < LT_W ? v * LIN_W : v;   // v_cmp_lt_f32 + v_mul_f32 + v_cndmask_b32
}

// Every block processes exactly BLOCK*UNROLL float4 packets; launch guarantees
// all accesses are in range -> no bounds checks, no exec masking.
__global__ __launch_bounds__(256) void ApplyLtLin_full_kernel(
    const float* __restrict__ x, float* __restrict__ y) {
    constexpr int BLOCK = 256;
    constexpr int UNROLL = 4;
    const v4f* __restrict__ xv = (const v4f*)x;
    v4f* __restrict__ yv = (v4f*)y;

    const int base = blockIdx.x * (BLOCK * UNROLL) + threadIdx.x;

    // 4 independent b128 NT loads in flight per lane.
    v4f v[UNROLL];
#pragma unroll
    for (int i = 0; i < UNROLL; ++i)
        v[i] = __builtin_nontemporal_load(&xv[base + i * BLOCK]);

#pragma unroll
    for (int i = 0; i < UNROLL; ++i) {
        v4f r = v[i];
#pragma unroll
        for (int c = 0; c < 4; ++c) r[c] = apply_elem(r[c]);
        __builtin_nontemporal_store(r, &yv[base + i * BLOCK]);
    }
}

// Scalar remainder pass over [start, n). Only launched when a remainder exists.
__global__ __launch_bounds__(256) void ApplyLtLin_tail_kernel(
    const float* __restrict__ x, float* __restrict__ y, int start, int n) {
    const int i = start + blockIdx.x * blockDim.x + threadIdx.x;
    if (i < n) y[i] = apply_elem(x[i]);
}

extern "C" void kernel_launch(void* const* d_in, const int* in_sizes, int n_in,
                              void* d_out, int out_size, void* d_ws, size_t ws_size,
                              hipStream_t stream) {
    (void)n_in; (void)d_ws; (void)ws_size; (void)out_size;
    const float* x = (const float*)d_in[0];
    float* y = (float*)d_out;
    const int n = in_sizes[0];

    constexpr int BLOCK = 256;                 // 8 wave32 waves per block
    constexpr int UNROLL = 4;
    const int elems_per_block = BLOCK * UNROLL * 4;   // 4096 floats per block

    const int full_blocks = n / elems_per_block;      // 16384 for n = 2^26
    const int done = full_blocks * elems_per_block;
    const int rem = n - done;                          // 0 for n = 2^26

    if (full_blocks > 0)
        ApplyLtLin_full_kernel<<<full_blocks, BLOCK, 0, stream>>>(x, y);

    if (rem > 0) {
        const int tail_blocks = (rem + BLOCK - 1) / BLOCK;
        ApplyLtLin_tail_kernel<<<tail_blocks, BLOCK, 0, stream>>>(x, y, done, n);
    }
}